// CGATNet_16853451670011
// MI455X (gfx1250) — compile-verified
//
#include <hip/hip_runtime.h>
#include <hip/hip_bf16.h>

#define HEADS 8
#define HID 16
#define NCLS 40
#define FIN 128
#define NEG_SLOPE 0.2f

typedef float v2f __attribute__((ext_vector_type(2)));
typedef float v8f __attribute__((ext_vector_type(8)));

// ---------------------------------------------------------------------------
// fp32 WMMA GEMM: C[M,N] = A[M,K] * B[K,N], all row-major.
// Requires M%16==0, N%16==0, K%4==0. One 16x16 output tile per wave,
// K-reduction via V_WMMA_F32_16X16X4_F32 (full fp32 precision).
// A 16x4 layout: lane(l<16) holds A[l][koff], A[l][koff+1]; l>=16 -> koff+2.
// B 4x16 layout mirrored (column per lane). C/D: VGPR i -> row i + 8*(l>=16).
// ---------------------------------------------------------------------------
__global__ void gemm_f32_wmma(const float* __restrict__ A,
                              const float* __restrict__ B,
                              float* __restrict__ C, int M, int N, int K) {
  const int lane  = threadIdx.x & 31;
  const int wave  = threadIdx.x >> 5;
  const int ntile = blockIdx.y * 4 + wave;
  if (ntile * 16 >= N) return;              // uniform per-wave exit; EXEC stays full
  const int m0   = blockIdx.x * 16;
  const int n0   = ntile * 16;
  const int r    = lane & 15;               // A row / B,C column within tile
  const int koff = (lane >> 4) << 1;        // 0 or 2

  v8f acc = {0.f, 0.f, 0.f, 0.f, 0.f, 0.f, 0.f, 0.f};
  const float* Arow = A + (size_t)(m0 + r) * K;
  for (int k = 0; k < K; k += 4) {
    v2f a = *(const v2f*)(Arow + k + koff);                 // 8B-aligned
    v2f b;
    b.x = B[(size_t)(k + koff)     * N + n0 + r];
    b.y = B[(size_t)(k + koff + 1) * N + n0 + r];
    acc = __builtin_amdgcn_wmma_f32_16x16x4_f32(
        false, a, false, b, (short)0, acc, false, false);
  }
  const int rbase = (lane >> 4) << 3;       // 0 or 8
  float* Cp = C + (size_t)(m0 + rbase) * N + n0 + r;
#pragma unroll
  for (int i = 0; i < 8; ++i) Cp[(size_t)i * N] = acc[i];
}

// ---------------------------------------------------------------------------
// Per-(node,head) attention coefficients: al = <h[n,h,:], a_src[h,:]>, etc.
// ---------------------------------------------------------------------------
__global__ void att_coef(const float* __restrict__ h,
                         const float* __restrict__ asr,
                         const float* __restrict__ adt,
                         float* __restrict__ al, float* __restrict__ ar,
                         int N, int C) {
  int t = blockIdx.x * blockDim.x + threadIdx.x;
  if (t >= N * HEADS) return;
  int n = t / HEADS, hh = t % HEADS;
  const float* hp = h + ((size_t)n * HEADS + hh) * C;
  const float* as = asr + hh * C;
  const float* ad = adt + hh * C;
  float sl = 0.f, sr = 0.f;
  for (int c = 0; c < C; ++c) { float v = hp[c]; sl += v * as[c]; sr += v * ad[c]; }
  al[t] = sl;
  ar[t] = sr;
}

// Monotonic float<->uint encoding so float segment-max can use atomicMax(u32).
__device__ __forceinline__ unsigned f2ord(float f) {
  unsigned u = __float_as_uint(f);
  return (u & 0x80000000u) ? ~u : (u | 0x80000000u);
}
__device__ __forceinline__ float ord2f(unsigned u) {
  return (u & 0x80000000u) ? __uint_as_float(u & 0x7fffffffu)
                           : __uint_as_float(~u);
}
#define ORD_NEG_INF 0x007FFFFFu   // f2ord(-inf)

__device__ __forceinline__ float leaky(float v) {
  return v > 0.f ? v : NEG_SLOPE * v;
}

__global__ void fill_u32(unsigned* __restrict__ p, unsigned v, int n) {
  int i = blockIdx.x * blockDim.x + threadIdx.x;
  if (i < n) p[i] = v;
}

// out[i] = bias[i % C]  (pre-fills d_out with b2 so layer2 scatters on top)
__global__ void fill_bias(float* __restrict__ out, const float* __restrict__ b,
                          int n, int C) {
  int i = blockIdx.x * blockDim.x + threadIdx.x;
  if (i < n) out[i] = b[i % C];
}

// ---------------------------------------------------------------------------
// Edge pass 1: segment max of leaky(al[src]+ar[dst]) into ordered-uint mb.
// edge_index is int64 per the reference (dtype=jnp.int64).
// ---------------------------------------------------------------------------
__global__ void edge_max(const long long* __restrict__ src,
                         const long long* __restrict__ dst,
                         const float* __restrict__ al,
                         const float* __restrict__ ar,
                         unsigned* __restrict__ mb, int E) {
  int e = blockIdx.x * blockDim.x + threadIdx.x;
  if (e >= E) return;
  int s = (int)src[e], d = (int)dst[e];
#pragma unroll
  for (int hh = 0; hh < HEADS; ++hh) {
    float v = leaky(al[s * HEADS + hh] + ar[d * HEADS + hh]);
    atomicMax(&mb[d * HEADS + hh], f2ord(v));
  }
}

// Edge pass 2: denom[dst,h] += exp(e - m[dst,h])
__global__ void edge_den(const long long* __restrict__ src,
                         const long long* __restrict__ dst,
                         const float* __restrict__ al,
                         const float* __restrict__ ar,
                         const unsigned* __restrict__ mb,
                         float* __restrict__ den, int E) {
  int e = blockIdx.x * blockDim.x + threadIdx.x;
  if (e >= E) return;
  int s = (int)src[e], d = (int)dst[e];
#pragma unroll
  for (int hh = 0; hh < HEADS; ++hh) {
    float v = leaky(al[s * HEADS + hh] + ar[d * HEADS + hh]);
    float m = ord2f(mb[d * HEADS + hh]);
    atomicAdd(&den[d * HEADS + hh], expf(v - m));
  }
}

// ---------------------------------------------------------------------------
// Edge pass 3: out[dst] += scale * alpha * feat[src].  One thread per
// (edge, head); alpha recomputed (cheap, cached) instead of a 102MB [E,H]
// temporary.  layer1: outLD=H*C, headStride=C, scale=1 (concat).
// layer2: outLD=C, headStride=0, scale=1/H (head mean fused into scatter).
// ---------------------------------------------------------------------------
__global__ void edge_scatter(const long long* __restrict__ src,
                             const long long* __restrict__ dst,
                             const float* __restrict__ al,
                             const float* __restrict__ ar,
                             const unsigned* __restrict__ mb,
                             const float* __restrict__ den,
                             const float* __restrict__ feat,
                             float* __restrict__ out,
                             int E, int C, int featLD, int outLD,
                             int outHeadStride, float scale) {
  long long t = (long long)blockIdx.x * blockDim.x + threadIdx.x;
  if (t >= (long long)E * HEADS) return;
  int e  = (int)(t / HEADS);
  int hh = (int)(t % HEADS);
  int s = (int)src[e], d = (int)dst[e];
  float v = leaky(al[s * HEADS + hh] + ar[d * HEADS + hh]);
  float m = ord2f(mb[d * HEADS + hh]);
  float alpha = expf(v - m) / den[d * HEADS + hh] * scale;
  const float* fp = feat + (size_t)s * featLD + hh * C;
  float* op = out + (size_t)d * outLD + hh * outHeadStride;
  for (int c = 0; c < C; ++c) atomicAdd(&op[c], alpha * fp[c]);
}

// h_out = elu(acc + b1)  (layer-1 output activation, written into h1 buffer)
__global__ void elu_bias(const float* __restrict__ acc,
                         const float* __restrict__ b,
                         float* __restrict__ out, int n, int C) {
  int i = blockIdx.x * blockDim.x + threadIdx.x;
  if (i >= n) return;
  float v = acc[i] + b[i % C];
  out[i] = v > 0.f ? v : expf(v) - 1.f;
}

extern "C" void kernel_launch(void* const* d_in, const int* in_sizes, int n_in,
                              void* d_out, int out_size, void* d_ws, size_t ws_size,
                              hipStream_t stream) {
  (void)n_in; (void)out_size; (void)ws_size;
  const float*     x   = (const float*)d_in[0];
  const long long* ei  = (const long long*)d_in[1];   // int64 per reference
  const float*     W1  = (const float*)d_in[2];
  const float*     as1 = (const float*)d_in[3];
  const float*     ad1 = (const float*)d_in[4];
  const float*     b1  = (const float*)d_in[5];
  const float*     W2  = (const float*)d_in[6];
  const float*     as2 = (const float*)d_in[7];
  const float*     ad2 = (const float*)d_in[8];
  const float*     b2  = (const float*)d_in[9];
  float* out = (float*)d_out;

  const int N = in_sizes[0] / FIN;   // 100000 (divisible by 16)
  const int E = in_sizes[1] / 2;     // 3200000
  const long long* srcI = ei;        // edge_index[0]
  const long long* dstI = ei + E;    // edge_index[1]

  // Workspace layout (~243 MB)
  char* w = (char*)d_ws;
  float*    h1   = (float*)w;    w += (size_t)N * FIN * 4;          // [N,128]
  float*    acc1 = (float*)w;    w += (size_t)N * FIN * 4;          // [N,128]
  float*    h2   = (float*)w;    w += (size_t)N * HEADS * NCLS * 4; // [N,320]
  float*    al   = (float*)w;    w += (size_t)N * HEADS * 4;
  float*    ar   = (float*)w;    w += (size_t)N * HEADS * 4;
  unsigned* mb   = (unsigned*)w; w += (size_t)N * HEADS * 4;
  float*    den  = (float*)w;    w += (size_t)N * HEADS * 4;

  const int NH = N * HEADS;
  dim3 blk(256);

  // ---------------- Layer 1 ----------------
  // h1 = x @ W1   (M=N, N=128, K=128)
  gemm_f32_wmma<<<dim3(N / 16, (FIN / 16 + 3) / 4), dim3(128), 0, stream>>>(
      x, W1, h1, N, FIN, FIN);
  att_coef<<<(NH + 255) / 256, blk, 0, stream>>>(h1, as1, ad1, al, ar, N, HID);

  fill_u32<<<(NH + 255) / 256, blk, 0, stream>>>(mb, ORD_NEG_INF, NH);
  hipMemsetAsync(den, 0, (size_t)NH * 4, stream);
  hipMemsetAsync(acc1, 0, (size_t)N * FIN * 4, stream);

  edge_max<<<(E + 255) / 256, blk, 0, stream>>>(srcI, dstI, al, ar, mb, E);
  edge_den<<<(E + 255) / 256, blk, 0, stream>>>(srcI, dstI, al, ar, mb, den, E);
  {
    long long tot = (long long)E * HEADS;
    edge_scatter<<<(unsigned)((tot + 255) / 256), blk, 0, stream>>>(
        srcI, dstI, al, ar, mb, den, h1, acc1,
        E, HID, HEADS * HID, HEADS * HID, HID, 1.0f);
  }
  // h1 <- elu(acc1 + b1)   (h1 buffer reused as layer-2 input)
  elu_bias<<<((N * FIN) + 255) / 256, blk, 0, stream>>>(acc1, b1, h1, N * FIN, FIN);

  // ---------------- Layer 2 ----------------
  // h2 = h1 @ W2   (M=N, N=320, K=128)
  gemm_f32_wmma<<<dim3(N / 16, (HEADS * NCLS / 16 + 3) / 4), dim3(128), 0, stream>>>(
      h1, W2, h2, N, HEADS * NCLS, FIN);
  att_coef<<<(NH + 255) / 256, blk, 0, stream>>>(h2, as2, ad2, al, ar, N, NCLS);

  fill_u32<<<(NH + 255) / 256, blk, 0, stream>>>(mb, ORD_NEG_INF, NH);
  hipMemsetAsync(den, 0, (size_t)NH * 4, stream);
  // d_out pre-filled with b2; head-mean (1/H) fused into scatter scale.
  fill_bias<<<((N * NCLS) + 255) / 256, blk, 0, stream>>>(out, b2, N * NCLS, NCLS);

  edge_max<<<(E + 255) / 256, blk, 0, stream>>>(srcI, dstI, al, ar, mb, E);
  edge_den<<<(E + 255) / 256, blk, 0, stream>>>(srcI, dstI, al, ar, mb, den, E);
  {
    long long tot = (long long)E * HEADS;
    edge_scatter<<<(unsigned)((tot + 255) / 256), blk, 0, stream>>>(
        srcI, dstI, al, ar, mb, den, h2, out,
        E, NCLS, HEADS * NCLS, NCLS, 0, 1.0f / HEADS);
  }
}